// MultiScaleBlockTPVDeformable_19731079757893
// MI455X (gfx1250) — compile-verified
//
#include <hip/hip_runtime.h>
#include <hip/hip_bf16.h>
#include <math.h>

typedef float v2f __attribute__((ext_vector_type(2)));
typedef float v8f __attribute__((ext_vector_type(8)));

#define ZDIM 16
#define HDIM 64
#define WDIM 64
#define VOX  (ZDIM*HDIM*WDIM)   // 65536
#define CCH  128

// ---------------------------------------------------------------------------
// Generic fp32 WMMA GEMM:  Out[m][n] (op)= epi( sum_k A[m][k]*B[k][n] + bias[n] )
//   A: row-major T x K (fp32), tile staged in LDS, rows >= T zero-padded.
//   B: arbitrary strides: element (k,n) at B[k*sK + n*sN].
//   One wave computes a 16x16 tile with K/4 V_WMMA_F32_16X16X4_F32 steps.
//   A 16x4 frag layout (ISA 7.12.2): lanes 0-15 -> K pair {0,1}, lanes 16-31 -> {2,3},
//   M = lane%16, VGPR0=first K, VGPR1=second K.  B 4x16 mirrors (row striped
//   across lanes per VGPR, half-split on lane 16).
// EPI: 0 = bias, 1 = bias + residual, 2 = accumulate into Out (no bias), 3 = gelu(bias)
// ---------------------------------------------------------------------------
template<int EPI>
__global__ void gemm_wmma_f32(const float* __restrict__ A, int T, int K,
                              const float* __restrict__ B, int sK, int sN,
                              const float* __restrict__ bias,
                              const float* __restrict__ resid,
                              float* __restrict__ Out, int ldOut)
{
    extern __shared__ float ldsA[];                  // 16 * K floats
    const int m0   = blockIdx.x * 16;
    const int lane = threadIdx.x;                    // 0..31
    const int wave = threadIdx.y;
    const int n0   = (blockIdx.y * blockDim.y + wave) * 16;

    // cooperative zero-padded load of the 16 x K A tile
    const int tid  = wave * 32 + lane;
    const int nthr = blockDim.y * 32;
    for (int idx = tid; idx < 16 * K; idx += nthr) {
        const int m = idx / K, k = idx - m * K;
        const int gm = m0 + m;
        ldsA[idx] = (gm < T) ? A[(size_t)gm * K + k] : 0.0f;
    }
    __syncthreads();

    const int half = lane >> 4;                      // 0: K{0,1}, 1: K{2,3}
    const int l16  = lane & 15;

    v8f acc = {0.f,0.f,0.f,0.f,0.f,0.f,0.f,0.f};
    for (int k0 = 0; k0 < K; k0 += 4) {
        const int ka = k0 + half * 2;
        v2f a, b;
        a.x = ldsA[l16 * K + ka];
        a.y = ldsA[l16 * K + ka + 1];
        b.x = B[(size_t)ka       * sK + (size_t)(n0 + l16) * sN];
        b.y = B[(size_t)(ka + 1) * sK + (size_t)(n0 + l16) * sN];
        acc = __builtin_amdgcn_wmma_f32_16x16x4_f32(false, a, false, b,
                                                    (short)0, acc, false, false);
    }

    const float bn = (EPI == 2) ? 0.0f : bias[n0 + l16];
    #pragma unroll
    for (int r = 0; r < 8; ++r) {
        const int m = m0 + r + half * 8;             // C/D layout: vgpr r, half-split
        if (m < T) {
            const size_t o = (size_t)m * ldOut + (n0 + l16);
            float v = acc[r] + bn;
            if (EPI == 1) v += resid[o];
            if (EPI == 3) v = 0.5f * v * (1.0f + erff(v * 0.70710678118654752f)); // exact GELU
            if (EPI == 2) Out[o] += v; else Out[o] = v;
        }
    }
}

// ---------------------------------------------------------------------------
// voxel v = zi*4096 + yi*64 + xi ; split s: block sizes bz,by,bx
// token t = bId*N + n, bId=(szi*s+syi)*s+sxi, n=(zl*by+yl)*bx+xl
// ---------------------------------------------------------------------------
__global__ void pack_kernel(const float* __restrict__ x, float* __restrict__ Xf,
                            int s, int bz, int by, int bx)
{
    const int t = blockIdx.x, c = threadIdx.x;
    const int N = bz * by * bx;
    const int b = t / N, n = t - b * N;
    const int szi = b / (s * s), syi = (b / s) % s, sxi = b % s;
    const int zl = n / (by * bx), r2 = n % (by * bx), yl = r2 / bx, xl = r2 % bx;
    const int v = (szi * bz + zl) * 4096 + (syi * by + yl) * 64 + (sxi * bx + xl);
    Xf[(size_t)t * CCH + c] = x[(size_t)c * VOX + v];
}

__global__ void softmax4_kernel(float* __restrict__ attn, int TH)
{
    const int g = blockIdx.x * blockDim.x + threadIdx.x;
    if (g >= TH) return;
    float* p = attn + (size_t)g * 4;
    const float m = fmaxf(fmaxf(p[0], p[1]), fmaxf(p[2], p[3]));
    const float e0 = expf(p[0]-m), e1 = expf(p[1]-m), e2 = expf(p[2]-m), e3 = expf(p[3]-m);
    const float inv = 1.0f / (e0 + e1 + e2 + e3);
    p[0] = e0*inv; p[1] = e1*inv; p[2] = e2*inv; p[3] = e3*inv;
}

// bilinear deformable sampling: block = 128 threads = (head h, channel dd) for one token
__global__ void sample_kernel(const float* __restrict__ value, const float* __restrict__ off,
                              const float* __restrict__ attn, float* __restrict__ S,
                              int N, int Hg, int Wg)
{
    const int t  = blockIdx.x;
    const int h  = threadIdx.x >> 4;
    const int dd = threadIdx.x & 15;
    const int b  = t / N;
    const int n  = t - b * N;
    const int row = n / Wg, col = n - row * Wg;
    const float* vb = value + (size_t)b * N * CCH + h * 16 + dd;
    const float* op = off  + (size_t)t * 64 + h * 8;
    const float* ap = attn + (size_t)t * 32 + h * 4;
    float acc = 0.0f;
    #pragma unroll
    for (int p = 0; p < 4; ++p) {
        const float ix = (float)col + op[p * 2 + 0];
        const float iy = (float)row + op[p * 2 + 1];
        const float fx0 = floorf(ix), fy0 = floorf(iy);
        const float wx = ix - fx0, wy = iy - fy0;
        const int x0 = (int)fx0, y0 = (int)fy0;
        const float a = ap[p];
        float c00 = 0.f, c10 = 0.f, c01 = 0.f, c11 = 0.f;
        if (x0   >= 0 && x0   < Wg && y0   >= 0 && y0   < Hg) c00 = vb[(size_t)(y0    *Wg + x0  ) * CCH];
        if (x0+1 >= 0 && x0+1 < Wg && y0   >= 0 && y0   < Hg) c10 = vb[(size_t)(y0    *Wg + x0+1) * CCH];
        if (x0   >= 0 && x0   < Wg && y0+1 >= 0 && y0+1 < Hg) c01 = vb[(size_t)((y0+1)*Wg + x0  ) * CCH];
        if (x0+1 >= 0 && x0+1 < Wg && y0+1 >= 0 && y0+1 < Hg) c11 = vb[(size_t)((y0+1)*Wg + x0+1) * CCH];
        acc += a * ((1.f-wx)*(1.f-wy)*c00 + wx*(1.f-wy)*c10 + (1.f-wx)*wy*c01 + wx*wy*c11);
    }
    S[(size_t)t * CCH + h * 16 + dd] = acc;
}

__global__ void block_mean_kernel(const float* __restrict__ X2, float* __restrict__ tok, int N)
{
    const int b = blockIdx.x, c = threadIdx.x;
    const float* base = X2 + (size_t)b * N * CCH + c;
    float sum = 0.0f;
    for (int n = 0; n < N; ++n) sum += base[(size_t)n * CCH];
    tok[(size_t)b * CCH + c] = sum / (float)N;
}

// out_s[v][c] = X2[t(v)][c] + tokX2[bId(v)][c] + x[c][v]   (intra + inter + residual)
__global__ void combine_kernel(const float* __restrict__ X2, const float* __restrict__ tokX2,
                               const float* __restrict__ x, float* __restrict__ out_s,
                               int s, int bz, int by, int bx)
{
    const int v = blockIdx.x, c = threadIdx.x;
    const int zi = v >> 12, rem = v & 4095, yi = rem >> 6, xi = rem & 63;
    const int szi = zi / bz, zl = zi % bz;
    const int syi = yi / by, yl = yi % by;
    const int sxi = xi / bx, xl = xi % bx;
    const int bId = (szi * s + syi) * s + sxi;
    const size_t t = (size_t)bId * (bz * by * bx) + (zl * by + yl) * bx + xl;
    out_s[(size_t)v * CCH + c] = X2[t * CCH + c] + tokX2[(size_t)bId * CCH + c]
                               + x[(size_t)c * VOX + v];
}

__global__ void final_kernel(const float* __restrict__ Y, const float* __restrict__ g,
                             const float* __restrict__ b, float* __restrict__ out)
{
    const int v = blockIdx.x, c = threadIdx.x;
    const float val = Y[(size_t)v * CCH + c] * g[c] + b[c];
    out[(size_t)c * VOX + v] = fmaxf(val, 0.0f);
}

// ---------------------------------------------------------------------------
static void launch_gemm(int epi, const float* A, int T, int K, const float* B,
                        int sK, int sN, const float* bias, const float* resid,
                        float* Out, int Nout, hipStream_t stream)
{
    int NW = Nout / 16; if (NW > 8) NW = 8;
    dim3 grid((T + 15) / 16, Nout / (16 * NW));
    dim3 block(32, NW);
    const size_t shmem = (size_t)16 * K * sizeof(float);
    switch (epi) {
        case 0: gemm_wmma_f32<0><<<grid, block, shmem, stream>>>(A,T,K,B,sK,sN,bias,resid,Out,Nout); break;
        case 1: gemm_wmma_f32<1><<<grid, block, shmem, stream>>>(A,T,K,B,sK,sN,bias,resid,Out,Nout); break;
        case 2: gemm_wmma_f32<2><<<grid, block, shmem, stream>>>(A,T,K,B,sK,sN,bias,resid,Out,Nout); break;
        case 3: gemm_wmma_f32<3><<<grid, block, shmem, stream>>>(A,T,K,B,sK,sN,bias,resid,Out,Nout); break;
    }
}

extern "C" void kernel_launch(void* const* d_in, const int* in_sizes, int n_in,
                              void* d_out, int out_size, void* d_ws, size_t ws_size,
                              hipStream_t stream)
{
    (void)in_sizes; (void)n_in; (void)out_size; (void)ws_size;
    const float* x    = (const float*)d_in[0];   // (1,128,16,64,64)
    const float* so_w = (const float*)d_in[1];   // (4,2,128,64)
    const float* so_b = (const float*)d_in[2];   // (4,2,64)
    const float* aw_w = (const float*)d_in[3];   // (4,2,128,32)
    const float* aw_b = (const float*)d_in[4];   // (4,2,32)
    const float* vp_w = (const float*)d_in[5];   // (4,2,128,128)
    const float* vp_b = (const float*)d_in[6];   // (4,2,128)
    const float* op_w = (const float*)d_in[7];   // (4,2,128,128)
    const float* op_b = (const float*)d_in[8];   // (4,2,128)
    const float* f1_w = (const float*)d_in[9];   // (4,256,128)
    const float* f1_b = (const float*)d_in[10];  // (4,256)
    const float* f2_w = (const float*)d_in[11];  // (4,128,256)
    const float* f2_b = (const float*)d_in[12];  // (4,128)
    const float* fuse_w = (const float*)d_in[13];// (128,512)
    const float* bn_g = (const float*)d_in[14];  // (128)
    const float* bn_b = (const float*)d_in[15];  // (128)
    float* out = (float*)d_out;

    // workspace carve-out (fp32)
    float* p = (float*)d_ws;
    float* Xf    = p; p += (size_t)VOX * 128;    // token matrix; reused as X2
    float* value = p; p += (size_t)VOX * 128;
    float* off   = p; p += (size_t)VOX * 64;
    float* attn  = p; p += (size_t)VOX * 32;
    float* S     = p; p += (size_t)VOX * 128;    // samples; reused as out_s
    float* X1    = p; p += (size_t)VOX * 128;
    float* Hbuf  = p; p += (size_t)VOX * 256;
    float* Y     = p; p += (size_t)VOX * 128;    // fused accumulator
    // inter-path (<=512 tokens)
    float* tXf   = p; p += 512 * 128;
    float* tVal  = p; p += 512 * 128;
    float* tOff  = p; p += 512 * 64;
    float* tAttn = p; p += 512 * 32;
    float* tS    = p; p += 512 * 128;
    float* tX1   = p; p += 512 * 128;
    float* tX2   = p; p += 512 * 128;
    float* tH    = p; p += 512 * 256;

    float* X2    = Xf;   // Xf dead after X1 is formed
    float* out_s = S;    // S dead after op-projection

    hipMemsetAsync(Y, 0, (size_t)VOX * 128 * sizeof(float), stream);

    const int splits[4] = {1, 2, 4, 8};
    for (int i = 0; i < 4; ++i) {
        const int s = splits[i];
        const int bz = ZDIM / s, by = HDIM / s, bx = WDIM / s;
        const int N = bz * by * bx, nb = s * s * s;
        const int T = VOX;
        // intra-block weights (r = 0)
        const float* vpw = vp_w + (size_t)(i*2+0)*128*128; const float* vpb = vp_b + (size_t)(i*2+0)*128;
        const float* sow = so_w + (size_t)(i*2+0)*128*64;  const float* sob = so_b + (size_t)(i*2+0)*64;
        const float* aww = aw_w + (size_t)(i*2+0)*128*32;  const float* awb = aw_b + (size_t)(i*2+0)*32;
        const float* opw = op_w + (size_t)(i*2+0)*128*128; const float* opb = op_b + (size_t)(i*2+0)*128;
        const float* w1  = f1_w + (size_t)i*256*128; const float* b1 = f1_b + (size_t)i*256;
        const float* w2  = f2_w + (size_t)i*128*256; const float* b2 = f2_b + (size_t)i*128;

        pack_kernel<<<T, 128, 0, stream>>>(x, Xf, s, bz, by, bx);
        launch_gemm(0, Xf, T, 128, vpw, 128, 1, vpb, nullptr, value, 128, stream);
        launch_gemm(0, Xf, T, 128, sow, 64, 1, sob, nullptr, off, 64, stream);
        launch_gemm(0, Xf, T, 128, aww, 32, 1, awb, nullptr, attn, 32, stream);
        softmax4_kernel<<<(T*8 + 255)/256, 256, 0, stream>>>(attn, T*8);
        sample_kernel<<<T, 128, 0, stream>>>(value, off, attn, S, N, bz*by, bx);
        launch_gemm(1, S, T, 128, opw, 128, 1, opb, Xf, X1, 128, stream);          // X1 = Xf + attn_out
        launch_gemm(3, X1, T, 128, w1, 1, 128, b1, nullptr, Hbuf, 256, stream);    // H = gelu(X1 w1 + b1)
        launch_gemm(1, Hbuf, T, 256, w2, 1, 256, b2, X1, X2, 128, stream);         // X2 = X1 + H w2 + b2

        // ---- inter-block path on block-mean tokens (r = 1) ----
        block_mean_kernel<<<nb, 128, 0, stream>>>(X2, tXf, N);
        const float* vpw2 = vp_w + (size_t)(i*2+1)*128*128; const float* vpb2 = vp_b + (size_t)(i*2+1)*128;
        const float* sow2 = so_w + (size_t)(i*2+1)*128*64;  const float* sob2 = so_b + (size_t)(i*2+1)*64;
        const float* aww2 = aw_w + (size_t)(i*2+1)*128*32;  const float* awb2 = aw_b + (size_t)(i*2+1)*32;
        const float* opw2 = op_w + (size_t)(i*2+1)*128*128; const float* opb2 = op_b + (size_t)(i*2+1)*128;
        const int T2 = nb;
        launch_gemm(0, tXf, T2, 128, vpw2, 128, 1, vpb2, nullptr, tVal, 128, stream);
        launch_gemm(0, tXf, T2, 128, sow2, 64, 1, sob2, nullptr, tOff, 64, stream);
        launch_gemm(0, tXf, T2, 128, aww2, 32, 1, awb2, nullptr, tAttn, 32, stream);
        softmax4_kernel<<<(T2*8 + 255)/256, 256, 0, stream>>>(tAttn, T2*8);
        sample_kernel<<<T2, 128, 0, stream>>>(tVal, tOff, tAttn, tS, T2, s*s, s);
        launch_gemm(1, tS, T2, 128, opw2, 128, 1, opb2, tXf, tX1, 128, stream);
        launch_gemm(3, tX1, T2, 128, w1, 1, 128, b1, nullptr, tH, 256, stream);
        launch_gemm(1, tH, T2, 256, w2, 1, 256, b2, tX1, tX2, 128, stream);

        // ---- combine: intra + inter + input residual, then fused accumulation ----
        combine_kernel<<<VOX, 128, 0, stream>>>(X2, tX2, x, out_s, s, bz, by, bx);
        // Y[v][c] += sum_k out_s[v][k] * fuse_w[c][i*128 + k]
        launch_gemm(2, out_s, VOX, 128, fuse_w + (size_t)i*128, 1, 512,
                    nullptr, nullptr, Y, 128, stream);
    }

    final_kernel<<<VOX, 128, 0, stream>>>(Y, bn_g, bn_b, out);
}